// GCN_73323681677856
// MI455X (gfx1250) — compile-verified
//
#include <hip/hip_runtime.h>
#include <hip/hip_bf16.h>

typedef __attribute__((ext_vector_type(2))) float v2f;
typedef __attribute__((ext_vector_type(8))) float v8f;
typedef __attribute__((ext_vector_type(4))) int   v4i;

#define NEG_SLOPE 0.01f

#if __has_builtin(__builtin_amdgcn_global_load_async_to_lds_b128) && \
    __has_builtin(__builtin_amdgcn_s_wait_asynccnt)
#define USE_ASYNC_LDS 1
#else
#define USE_ASYNC_LDS 0
#endif

// ---------------------------------------------------------------------------
// Degree / normalization kernels:  A_hat = D^-1/2 (A + I) D^-1/2
// ---------------------------------------------------------------------------
__global__ void deg_init_kernel(float* __restrict__ deg, int n) {
  int i = blockIdx.x * blockDim.x + threadIdx.x;
  if (i < n) deg[i] = 1.0f;  // self-loop contributes weight 1 at dst
}

__global__ void deg_acc_kernel(float* __restrict__ deg, const long long* __restrict__ dst,
                               const float* __restrict__ ew, long long E) {
  long long i = (long long)blockIdx.x * blockDim.x + threadIdx.x;
  long long stride = (long long)gridDim.x * blockDim.x;
  for (long long e = i; e < E; e += stride)
    atomicAdd(&deg[dst[e]], ew[e]);
}

__global__ void dis_kernel(float* __restrict__ deg, int n) {
  int i = blockIdx.x * blockDim.x + threadIdx.x;
  if (i < n) {
    float d = deg[i];
    deg[i] = (d > 0.0f) ? (1.0f / sqrtf(d)) : 0.0f;  // in-place -> dis
  }
}

// ---------------------------------------------------------------------------
// fp32 WMMA GEMM:  out[M,NOUT] = act(A[M,K] @ W[K,NOUT] (+ bias))
// Block = 128 threads = 4 waves (wave32). Each wave owns a 16-row strip and
// NT 16-col tiles. A tile (64 x K) staged in LDS (async-to-LDS on gfx1250);
// W streams from L2 (<=32KB).
//
// V_WMMA_F32_16X16X4_F32 fragment layouts (ISA 7.12.2, wave32):
//   A 16x4 : lane l, vgpr v  -> A[l%16][v + 2*(l/16)]
//   B 4x16 : lane l, vgpr v  -> B[v + 2*(l/16)][l%16]
//   C/D    : lane l, vgpr r  -> C[r + 8*(l/16)][l%16]
// ---------------------------------------------------------------------------
template <int K, int NT, int NOUT, bool ACT, bool BIAS>
__global__ __launch_bounds__(128) void wmma_gemm_kernel(
    const float* __restrict__ A, const float* __restrict__ W,
    const float* __restrict__ bias, float* __restrict__ out, int n) {
  constexpr int K4 = K / 4;
  __shared__ __align__(16) float sA[64 * K];

  const int t    = threadIdx.x;
  const int lane = t & 31;
  const int wave = t >> 5;
  const int hi   = lane >> 4;   // which half-wave (K / M-row split)
  const int lm   = lane & 15;
  const int row0 = blockIdx.x * 64;

  // Stage 64 x K tile of A into LDS, float4 granularity, zero-padded tail rows.
  const float4* A4 = reinterpret_cast<const float4*>(A);
  float4* s4 = reinterpret_cast<float4*>(sA);
#if USE_ASYNC_LDS
  for (int i = t; i < 16 * K; i += 128) {
    int r = i / K4;
    int c = i - r * K4;
    if (row0 + r < n) {
      __builtin_amdgcn_global_load_async_to_lds_b128(
          (__attribute__((address_space(1))) v4i*)(A4 + (long long)(row0 + r) * K4 + c),
          (__attribute__((address_space(3))) v4i*)(s4 + i),
          0, 0);
    } else {
      s4[i] = make_float4(0.f, 0.f, 0.f, 0.f);
    }
  }
  __builtin_amdgcn_s_wait_asynccnt(0);  // drain this wave's async copies
  __syncthreads();                      // make all waves' tiles visible
#else
  for (int i = t; i < 16 * K; i += 128) {
    int r = i / K4;
    int c = i - r * K4;
    float4 v = make_float4(0.f, 0.f, 0.f, 0.f);
    if (row0 + r < n) v = A4[(long long)(row0 + r) * K4 + c];
    s4[i] = v;
  }
  __syncthreads();
#endif

  const float* ar = sA + (wave * 16 + lm) * K;  // this lane's A row (m = lm)
  v8f acc[NT] = {};

  for (int kk = 0; kk < K; kk += 4) {
    v2f a;
    a.x = ar[kk + 2 * hi];       // vgpr0: K = kk + 2*hi
    a.y = ar[kk + 2 * hi + 1];   // vgpr1: K = kk + 2*hi + 1
#pragma unroll
    for (int q = 0; q < NT; ++q) {
      const int c = q * 16 + lm;
      v2f b;
      if constexpr (NT * 16 == NOUT) {
        b.x = W[(long long)(kk + 2 * hi) * NOUT + c];
        b.y = W[(long long)(kk + 2 * hi + 1) * NOUT + c];
      } else {
        b.x = (c < NOUT) ? W[(long long)(kk + 2 * hi) * NOUT + c] : 0.0f;
        b.y = (c < NOUT) ? W[(long long)(kk + 2 * hi + 1) * NOUT + c] : 0.0f;
      }
      acc[q] = __builtin_amdgcn_wmma_f32_16x16x4_f32(
          false, a, false, b, (short)0, acc[q], false, false);
    }
  }

  const int mbase = row0 + wave * 16 + 8 * hi;
  if (row0 + 64 <= n) {
    // Fast path: whole tile in range -> unguarded stores (all but tail block).
#pragma unroll
    for (int q = 0; q < NT; ++q) {
      const int c = q * 16 + lm;
      if constexpr (NT * 16 != NOUT) {
        if (c >= NOUT) continue;
      }
      float bv = 0.0f;
      if constexpr (BIAS) bv = bias[c];
#pragma unroll
      for (int r = 0; r < 8; ++r) {
        float v = acc[q][r];
        if constexpr (BIAS) v += bv;
        if (ACT) v = (v >= 0.0f) ? v : NEG_SLOPE * v;
        out[(long long)(mbase + r) * NOUT + c] = v;
      }
    }
  } else {
#pragma unroll
    for (int q = 0; q < NT; ++q) {
      const int c = q * 16 + lm;
#pragma unroll
      for (int r = 0; r < 8; ++r) {
        const int m = mbase + r;
        if (m < n && c < NOUT) {
          float v = acc[q][r];
          if (BIAS) v += bias[c];
          if (ACT) v = (v >= 0.0f) ? v : NEG_SLOPE * v;
          out[(long long)m * NOUT + c] = v;
        }
      }
    }
  }
}

// ---------------------------------------------------------------------------
// Aggregation: agg[i][:] = bias + dis[i]^2 * t[i][:]   (self-loop term)
// then edge scatter: agg[dst] += (dis[src]*w*dis[dst]) * t[src]
// ---------------------------------------------------------------------------
__global__ void scatter_init_kernel(const float* __restrict__ t, const float* __restrict__ dis,
                                    const float* __restrict__ bias, float* __restrict__ agg,
                                    long long total) {
  long long i = (long long)blockIdx.x * blockDim.x + threadIdx.x;
  if (i < total) {
    long long node = i >> 6;
    int j = (int)(i & 63);
    float d = dis[node];
    agg[i] = bias[j] + d * d * t[i];
  }
}

__global__ void scatter_edges_kernel(const float* __restrict__ t, const float* __restrict__ dis,
                                     const long long* __restrict__ src,
                                     const long long* __restrict__ dst,
                                     const float* __restrict__ ew, float* __restrict__ agg,
                                     long long E) {
  const int lane = threadIdx.x & 31;
  long long w  = ((long long)blockIdx.x * blockDim.x + threadIdx.x) >> 5;
  long long nw = ((long long)gridDim.x * blockDim.x) >> 5;
  for (long long e = w; e < E; e += nw) {  // one wave32 per edge: 64 feats, 2/lane
    long long s = src[e], d = dst[e];
    float nrm = dis[s] * ew[e] * dis[d];
    const float* ts = t + s * 64;
    float* ad = agg + d * 64;
    atomicAdd(&ad[lane],      nrm * ts[lane]);
    atomicAdd(&ad[lane + 32], nrm * ts[lane + 32]);
  }
}

__global__ void lrelu_kernel(float* __restrict__ h, long long total) {
  long long i = (long long)blockIdx.x * blockDim.x + threadIdx.x;
  if (i < total) {
    float v = h[i];
    h[i] = (v >= 0.0f) ? v : NEG_SLOPE * v;
  }
}

__global__ void log_softmax_kernel(const float* __restrict__ logits, float* __restrict__ out, int n) {
  int i = blockIdx.x * blockDim.x + threadIdx.x;
  if (i < n) {
    const float* l = logits + (long long)i * 40;
    float m = l[0];
    for (int j = 1; j < 40; ++j) m = fmaxf(m, l[j]);
    float s = 0.0f;
    for (int j = 0; j < 40; ++j) s += expf(l[j] - m);
    float ls = logf(s);
    float* o = out + (long long)i * 40;
    for (int j = 0; j < 40; ++j) o[j] = l[j] - m - ls;
  }
}

// ---------------------------------------------------------------------------
extern "C" void kernel_launch(void* const* d_in, const int* in_sizes, int n_in,
                              void* d_out, int out_size, void* d_ws, size_t ws_size,
                              hipStream_t stream) {
  const float*     x   = (const float*)d_in[0];
  const long long* ei  = (const long long*)d_in[1];  // [2, E] int64
  const float*     ew  = (const float*)d_in[2];
  const float*     Wf  = (const float*)d_in[3];
  const float*     bf  = (const float*)d_in[4];
  const float*     Wc1 = (const float*)d_in[5];
  const float*     bc1 = (const float*)d_in[6];
  const float*     Wc2 = (const float*)d_in[7];
  const float*     bc2 = (const float*)d_in[8];
  const float*     Wo  = (const float*)d_in[9];
  const float*     bo  = (const float*)d_in[10];
  float* out = (float*)d_out;

  const int N = in_sizes[0] / 128;
  const long long E = in_sizes[2];
  const long long* src = ei;
  const long long* dst = ei + E;

  float* ws = (float*)d_ws;
  long long NA = ((long long)N + 255) & ~255LL;
  float* dis  = ws;                       // [N]
  float* bufA = ws + NA;                  // [N,64]
  float* bufB = bufA + (long long)N * 64; // [N,64]

  const int gN = (N + 255) / 256;
  const long long total = (long long)N * 64;
  const int gT = (int)((total + 255) / 256);
  const int gG = (N + 63) / 64;

  // normalization: deg = 1 + sum_e w[dst], dis = deg^-1/2
  deg_init_kernel<<<gN, 256, 0, stream>>>(dis, N);
  deg_acc_kernel<<<2048, 256, 0, stream>>>(dis, dst, ew, E);
  dis_kernel<<<gN, 256, 0, stream>>>(dis, N);

  // h = lrelu(x @ W_first + b_first)
  wmma_gemm_kernel<128, 4, 64, true, true><<<gG, 128, 0, stream>>>(x, Wf, bf, bufA, N);

  // conv layer 1
  wmma_gemm_kernel<64, 4, 64, false, false><<<gG, 128, 0, stream>>>(bufA, Wc1, nullptr, bufB, N);
  scatter_init_kernel<<<gT, 256, 0, stream>>>(bufB, dis, bc1, bufA, total);
  scatter_edges_kernel<<<4096, 256, 0, stream>>>(bufB, dis, src, dst, ew, bufA, E);
  lrelu_kernel<<<gT, 256, 0, stream>>>(bufA, total);

  // conv layer 2
  wmma_gemm_kernel<64, 4, 64, false, false><<<gG, 128, 0, stream>>>(bufA, Wc2, nullptr, bufB, N);
  scatter_init_kernel<<<gT, 256, 0, stream>>>(bufB, dis, bc2, bufA, total);
  scatter_edges_kernel<<<4096, 256, 0, stream>>>(bufB, dis, src, dst, ew, bufA, E);
  lrelu_kernel<<<gT, 256, 0, stream>>>(bufA, total);

  // logits = h @ W_out + b_out ; log_softmax
  wmma_gemm_kernel<64, 3, 40, false, true><<<gG, 128, 0, stream>>>(bufA, Wo, bo, bufB, N);
  log_softmax_kernel<<<gN, 256, 0, stream>>>(bufB, out, N);

  (void)n_in; (void)out_size; (void)ws_size;
}